// SpikingLayer_5935644803784
// MI455X (gfx1250) — compile-verified
//
#include <hip/hip_runtime.h>
#include <stdint.h>

// SpikingLayer temporal scan, CDNA5 (gfx1250).
// x: (B*T, 32,32,32) f32, B=16, T=128. Scan over T per feature element:
//   state = syn + state - act; state = max(state,-1); act = max(floor(state),0)
// Memory-bound: 512 MiB traffic -> ~23 us floor @ 23.3 TB/s.
// Uses async global->LDS prefetch pipeline (ASYNCcnt) to hide HBM latency
// across the serial scan dependence, + non-temporal 128-bit stores.

typedef float v4f __attribute__((ext_vector_type(4)));

#define T_STEPS   128
#define VEC_PER_T 8192          // (32*32*32)/4 float4 per (b,t) slice
#define BLK       256           // 8 waves of 32
#define DEPTH     16            // async prefetch stages (64 KiB LDS / block)

__global__ __launch_bounds__(BLK)
void spiking_scan_kernel(const float* __restrict__ xin, float* __restrict__ xout) {
    __shared__ v4f buf[DEPTH * BLK];   // 64 KiB; slot s of thread tid: buf[s*BLK + tid]

    const int col = blockIdx.x * BLK + threadIdx.x;   // 0..131071
    const int b   = col >> 13;                        // 8192 float4-columns per batch
    const int f4  = col & (VEC_PER_T - 1);

    const size_t base = (size_t)b * ((size_t)T_STEPS * VEC_PER_T) + (size_t)f4;
    const v4f* gin  = (const v4f*)xin  + base;
    v4f*       gout = (v4f*)      xout + base;

    // Low 32 bits of the generic (addrspace-cast) shared pointer are the LDS
    // byte address.
    const unsigned lds_base = (unsigned)(uintptr_t)&buf[threadIdx.x];

    // ---- prologue: fill the async pipeline with DEPTH stages ----
    #pragma unroll
    for (int s = 0; s < DEPTH; ++s) {
        unsigned long long ga = (unsigned long long)(uintptr_t)(gin + (size_t)s * VEC_PER_T);
        unsigned la = lds_base + s * (BLK * (int)sizeof(v4f));
        asm volatile("global_load_async_to_lds_b128 %0, %1, off"
                     :: "v"(la), "v"(ga) : "memory");
    }

    v4f state = {0.f, 0.f, 0.f, 0.f};
    v4f act   = {0.f, 0.f, 0.f, 0.f};
    const v4f neg1 = {-1.f, -1.f, -1.f, -1.f};
    const v4f zero = { 0.f,  0.f,  0.f,  0.f};

    int t = 0;
    // ---- steady state: consume stage t, refill with stage t+DEPTH ----
    for (; t < T_STEPS - DEPTH; ++t) {
        // Async loads complete in order: cnt <= DEPTH-1 => stage t has landed.
        asm volatile("s_wait_asynccnt %0" :: "i"(DEPTH - 1) : "memory");

        const int slot = t & (DEPTH - 1);
        v4f syn = buf[slot * BLK + threadIdx.x];           // ds_load_b128

        state = state + syn - act;
        state = __builtin_elementwise_max(state, neg1);                       // clamp at th_low
        act   = __builtin_elementwise_max(__builtin_elementwise_floor(state), // spikes
                                          zero);
        __builtin_nontemporal_store(act, gout + (size_t)t * VEC_PER_T);       // b128 NT store

        // Reuse this slot for stage t+DEPTH. s_wait_dscnt 0 guarantees our
        // ds_load above fully read the slot before the async engine rewrites it.
        unsigned long long ga =
            (unsigned long long)(uintptr_t)(gin + (size_t)(t + DEPTH) * VEC_PER_T);
        unsigned la = lds_base + slot * (BLK * (int)sizeof(v4f));
        asm volatile("s_wait_dscnt 0\n\t"
                     "global_load_async_to_lds_b128 %0, %1, off"
                     :: "v"(la), "v"(ga) : "memory");
    }

    // ---- epilogue: drain remaining DEPTH stages ----
    asm volatile("s_wait_asynccnt 0" ::: "memory");
    for (; t < T_STEPS; ++t) {
        const int slot = t & (DEPTH - 1);
        v4f syn = buf[slot * BLK + threadIdx.x];
        state = state + syn - act;
        state = __builtin_elementwise_max(state, neg1);
        act   = __builtin_elementwise_max(__builtin_elementwise_floor(state), zero);
        __builtin_nontemporal_store(act, gout + (size_t)t * VEC_PER_T);
    }
}

extern "C" void kernel_launch(void* const* d_in, const int* in_sizes, int n_in,
                              void* d_out, int out_size, void* d_ws, size_t ws_size,
                              hipStream_t stream) {
    (void)n_in; (void)d_ws; (void)ws_size; (void)out_size;
    const float* x   = (const float*)d_in[0];
    float*       out = (float*)d_out;

    // total columns = n / T / 4 ; reference: 67,108,864 / 128 / 4 = 131,072
    const int total_vec_cols = in_sizes[0] / T_STEPS / 4;
    const int grid = total_vec_cols / BLK;               // 512 blocks

    spiking_scan_kernel<<<grid, BLK, 0, stream>>>(x, out);
}